// Actor_54262616818358
// MI455X (gfx1250) — compile-verified
//
#include <hip/hip_runtime.h>
#include <hip/hip_bf16.h>

#define NN 4096
#define DD 32
#define PP 8
#define EPS 1e-8f

typedef float v2f __attribute__((ext_vector_type(2)));
typedef float v8f __attribute__((ext_vector_type(8)));

#if __has_builtin(__builtin_amdgcn_tanhf)
static __device__ __forceinline__ float fast_tanh(float x) { return __builtin_amdgcn_tanhf(x); }
#else
static __device__ __forceinline__ float fast_tanh(float x) { return tanhf(x); }
#endif

// ---------------------------------------------------------------------------
// 16x16 Gram tile via V_WMMA_F32_16X16X4_F32:  C[m][n] = sum_k f[i0+m][k]*f[j0+n][k]
// A (16x4 f32): lane = M, VGPR0 = K+0 (lanes0-15) / K+2 (lanes16-31), VGPR1 = K+1/K+3.
// B (4x16 f32): mirrored with lane = N.
// C/D (16x16 f32, 8 VGPRs): m = v + 8*(lane>=16), n = lane&15.
// ---------------------------------------------------------------------------
__device__ __forceinline__ v8f gram_tile(const float* __restrict__ f,
                                         int i0, int j0, int lane) {
    const int mn   = lane & 15;
    const int koff = (lane >> 4) << 1;            // 0 or 2
    const float* __restrict__ ai = f + (size_t)(i0 + mn) * DD + koff;
    const float* __restrict__ bj = f + (size_t)(j0 + mn) * DD + koff;
    v8f c = {0.f, 0.f, 0.f, 0.f, 0.f, 0.f, 0.f, 0.f};
#pragma unroll
    for (int k = 0; k < DD; k += 4) {
        v2f a; a.x = ai[k]; a.y = ai[k + 1];
        v2f b; b.x = bj[k]; b.y = bj[k + 1];
        c = __builtin_amdgcn_wmma_f32_16x16x4_f32(false, a, false, b,
                                                  (short)0, c, false, false);
    }
    return c;
}

// ---------------------------------------------------------------------------
// Kernel 1a: row-normalize attributes into attr_n.  One wave per row (D==32).
// ---------------------------------------------------------------------------
__global__ void k_norm(const float* __restrict__ attr, float* __restrict__ attr_n) {
    const int lane = threadIdx.x & 31;
    const int row  = blockIdx.x * 8 + (threadIdx.x >> 5);
    float v = attr[(size_t)row * DD + lane];
    float s = v * v;
#pragma unroll
    for (int off = 16; off >= 1; off >>= 1) s += __shfl_xor(s, off, 32);
    attr_n[(size_t)row * DD + lane] = v / (sqrtf(s) + EPS);
}

// ---------------------------------------------------------------------------
// Kernel 1b: agg = binarize(edges) @ attr_n via WMMA.
// Block = 8 waves, one 16-row tile; waves split K (512 each), A fragment is
// binarized edges, two C tiles cover the 32 output columns; LDS tree-reduce.
// ---------------------------------------------------------------------------
__global__ void k_agg(const float* __restrict__ edges,
                      const float* __restrict__ attr_n,
                      float* __restrict__ agg) {
    const int lane = threadIdx.x & 31;
    const int wave = threadIdx.x >> 5;            // 0..7
    const int i0   = blockIdx.x * 16;
    const int m    = lane & 15;
    const int koff = (lane >> 4) << 1;            // 0 or 2

    v8f c0 = {0.f, 0.f, 0.f, 0.f, 0.f, 0.f, 0.f, 0.f};
    v8f c1 = {0.f, 0.f, 0.f, 0.f, 0.f, 0.f, 0.f, 0.f};
    const float* __restrict__ erow = edges + (size_t)(i0 + m) * NN + koff;
    const int kbase = wave * (NN / 8);
#pragma unroll 4
    for (int kk = 0; kk < NN / 8; kk += 4) {
        const int k = kbase + kk;
        v2f a;
        a.x = (erow[k]     > 0.f) ? 1.f : 0.f;
        a.y = (erow[k + 1] > 0.f) ? 1.f : 0.f;
        const float* __restrict__ bp = attr_n + (size_t)(k + koff) * DD;
        v2f b0; b0.x = bp[m];      b0.y = bp[DD + m];
        v2f b1; b1.x = bp[16 + m]; b1.y = bp[DD + 16 + m];
        c0 = __builtin_amdgcn_wmma_f32_16x16x4_f32(false, a, false, b0,
                                                   (short)0, c0, false, false);
        c1 = __builtin_amdgcn_wmma_f32_16x16x4_f32(false, a, false, b1,
                                                   (short)0, c1, false, false);
    }

    __shared__ float red[8][16 * DD];
#pragma unroll
    for (int v = 0; v < 8; ++v) {
        const int row = v + ((lane >> 4) << 3);
        red[wave][row * DD + m]      = c0[v];
        red[wave][row * DD + 16 + m] = c1[v];
    }
    __syncthreads();
    for (int t = threadIdx.x; t < 16 * DD; t += 256) {
        float s = 0.f;
#pragma unroll
        for (int w = 0; w < 8; ++w) s += red[w][t];
        agg[(size_t)i0 * DD + t] = s;
    }
}

// ---------------------------------------------------------------------------
// Kernel 1c: blend feats for all 8 personas; emit feat of persona 7 with f > 0 set to 1.
// ---------------------------------------------------------------------------
__global__ void k_feat(const float* __restrict__ attr, const float* __restrict__ agg,
                       const float* __restrict__ r, const float* __restrict__ W,
                       float* __restrict__ feats, float* __restrict__ feat_bin_out) {
    const size_t tid = (size_t)blockIdx.x * blockDim.x + threadIdx.x; // over N*D
    const float a = attr[tid];
    const float g = agg[tid];
#pragma unroll
    for (int p = 0; p < PP; ++p) {
        const float rp = r[p] + EPS;
        const float f  = rp * a + (W[p] * g) * (1.f - rp);
        feats[(size_t)p * NN * DD + tid] = f;
        if (p == PP - 1) feat_bin_out[tid] = (f > 0.f) ? 1.f : f;
    }
}

// ---------------------------------------------------------------------------
// Pass 1: per-persona column min/max of the Gram, folded through the monotone
// map y = e * exp(min(G/T + eps, 78)).  Block = 4 waves on one 16-col tile.
// ---------------------------------------------------------------------------
__global__ void k_pass1(const float* __restrict__ feats,
                        const float* __restrict__ T, const float* __restrict__ e,
                        float* __restrict__ colmn, float* __restrict__ colmx) {
    const int lane = threadIdx.x & 31;
    const int wave = threadIdx.x >> 5;        // 0..3
    const int p    = blockIdx.y;
    const int j0   = blockIdx.x * 16;
    const float* __restrict__ f = feats + (size_t)p * NN * DD;

    float gmn = 3.4e38f, gmx = -3.4e38f;
    for (int it = wave; it < NN / 16; it += 4) {
        v8f c = gram_tile(f, it * 16, j0, lane);
#pragma unroll
        for (int v = 0; v < 8; ++v) {
            gmn = fminf(gmn, c[v]);
            gmx = fmaxf(gmx, c[v]);
        }
    }
    gmn = fminf(gmn, __shfl_xor(gmn, 16, 32));
    gmx = fmaxf(gmx, __shfl_xor(gmx, 16, 32));

    __shared__ float smn[4][16];
    __shared__ float smx[4][16];
    if (lane < 16) { smn[wave][lane] = gmn; smx[wave][lane] = gmx; }
    __syncthreads();
    if (wave == 0 && lane < 16) {
        float a = fminf(fminf(smn[0][lane], smn[1][lane]),
                        fminf(smn[2][lane], smn[3][lane]));
        float b = fmaxf(fmaxf(smx[0][lane], smx[1][lane]),
                        fmaxf(smx[2][lane], smx[3][lane]));
        const float rT = 1.f / T[p];
        const float ei = e[p];
        colmn[p * NN + j0 + lane] = ei * __expf(fminf(a * rT + EPS, 78.f));
        colmx[p * NN + j0 + lane] = ei * __expf(fminf(b * rT + EPS, 78.f));
    }
}

// ---------------------------------------------------------------------------
// Pass 2: recompute each 16x16 Gram tile for all 8 personas; prob tile lives
// in registers across personas and is written exactly once.
// ---------------------------------------------------------------------------
__global__ void k_pass2(const float* __restrict__ feats,
                        const float* __restrict__ T, const float* __restrict__ e,
                        const float* __restrict__ persona,
                        const float* __restrict__ colmn, const float* __restrict__ colmx,
                        float* __restrict__ prob) {
    const int lane = threadIdx.x & 31;
    const int wave = threadIdx.x >> 5;                 // 0..7
    const int i0   = blockIdx.x * 16;
    const int j0   = (blockIdx.y * 8 + wave) * 16;
    const int n    = lane & 15;

    float acc[8] = {0.f, 0.f, 0.f, 0.f, 0.f, 0.f, 0.f, 0.f};
#pragma unroll
    for (int p = 0; p < PP; ++p) {
        const float* __restrict__ f = feats + (size_t)p * NN * DD;
        v8f c = gram_tile(f, i0, j0, lane);
        const float rT  = 1.f / T[p];
        const float ei  = e[p];
        const float mn  = colmn[p * NN + j0 + n];
        const float mx  = colmx[p * NN + j0 + n];
        const float pj  = persona[(size_t)(j0 + n) * PP + p];
        const float inv = 1.f / (mx - mn + EPS);
#pragma unroll
        for (int v = 0; v < 8; ++v) {
            float g = fminf(c[v] * rT + EPS, 78.f);
            float y = __expf(g) * ei;
            float t = fast_tanh((y - mn) * inv);
            acc[v] = fminf(fmaxf(acc[v] + pj * t, 0.f), 1.f);
        }
    }
#pragma unroll
    for (int v = 0; v < 8; ++v) {
        int m = v + ((lane >> 4) << 3);
        prob[(size_t)(i0 + m) * NN + j0 + n] = acc[v];
    }
}

// ---------------------------------------------------------------------------
extern "C" void kernel_launch(void* const* d_in, const int* in_sizes, int n_in,
                              void* d_out, int out_size, void* d_ws, size_t ws_size,
                              hipStream_t stream) {
    const float* attr    = (const float*)d_in[0];
    const float* edges   = (const float*)d_in[1];
    const float* T       = (const float*)d_in[2];
    const float* e       = (const float*)d_in[3];
    const float* r       = (const float*)d_in[4];
    const float* W       = (const float*)d_in[5];
    const float* persona = (const float*)d_in[6];

    float* prob     = (float*)d_out;                       // N*N
    float* feat_bin = (float*)d_out + (size_t)NN * NN;     // N*D

    float* ws      = (float*)d_ws;
    float* attr_n  = ws;                                   // N*D
    float* agg     = attr_n + (size_t)NN * DD;             // N*D
    float* feats   = agg    + (size_t)NN * DD;             // P*N*D
    float* colmn   = feats  + (size_t)PP * NN * DD;        // P*N
    float* colmx   = colmn  + (size_t)PP * NN;             // P*N

    k_norm<<<NN / 8, 256, 0, stream>>>(attr, attr_n);
    k_agg <<<NN / 16, 256, 0, stream>>>(edges, attr_n, agg);
    k_feat<<<(NN * DD) / 256, 256, 0, stream>>>(attr, agg, r, W, feats, feat_bin);
    k_pass1<<<dim3(NN / 16, PP), 128, 0, stream>>>(feats, T, e, colmn, colmx);
    k_pass2<<<dim3(NN / 16, NN / 128), 256, 0, stream>>>(feats, T, e, persona,
                                                         colmn, colmx, prob);
}